// Decoder_55095840473198
// MI455X (gfx1250) — compile-verified
//
#include <hip/hip_runtime.h>

typedef __attribute__((ext_vector_type(2))) float v2f;
typedef __attribute__((ext_vector_type(4))) float v4f;
typedef __attribute__((ext_vector_type(8))) float v8f;

#define NSLICES 16
#define WPB 8 // waves per block (wave32)

// Per-prefetch pointer strides (in floats)
#define DSTRIDE ((size_t)NSLICES * 16 * 64) // dataset: 16 rows x 64 cols per chunk
#define ASTRIDE ((size_t)NSLICES * 16 * 16) // alpha:   16 rows x 16 cols per chunk
#define XSTRIDE ((size_t)NSLICES * 16)      // xsq:     16 norms per chunk

#define LOG2E 1.4426950408889634f
#define NHALF_LOG2E (-0.7213475204444817f) // -0.5 * log2(e)

// Single-instruction hardware 2^x (v_exp_f32); avoids OCML exp2f's denorm fixups.
extern "C" __device__ float __ocml_native_exp2_f32(float);
#if __has_builtin(__builtin_amdgcn_exp2f)
#define EXP2F(x) __builtin_amdgcn_exp2f(x)
#else
#define EXP2F(x) __ocml_native_exp2_f32(x)
#endif

__global__ __launch_bounds__(256) void rbf_zero(float* p, int n) {
  int i = blockIdx.x * blockDim.x + threadIdx.x;
  if (i < n) p[i] = 0.0f;
}

// Precompute ||dataset[n]||^2 once (otherwise recomputed by all 128 m-tile waves).
__global__ __launch_bounds__(256) void rbf_xsq(const float* __restrict__ ds,
                                               float* __restrict__ xsq, int N) {
  int n = blockIdx.x * blockDim.x + threadIdx.x;
  if (n >= N) return;
  const v4f* r = (const v4f*)(ds + (size_t)n * 64);
  float s = 0.0f;
#pragma unroll
  for (int i = 0; i < 16; ++i) {
    v4f q = r[i];
    s = fmaf(q.x, q.x, fmaf(q.y, q.y, fmaf(q.z, q.z, fmaf(q.w, q.w, s))));
  }
  xsq[n] = s;
}

// Prefetch one 16-point chunk off persistent pointers with immediate offsets,
// then advance pointers by a scalar-selected stride (0 for the tail dummy).
template <bool PRE>
__device__ __forceinline__ void load_chunk(const float*& dptr, const float*& aptr,
                                           const float*& xptr, bool advance,
                                           v2f (&at)[16], v2f (&aa)[4], v4f (&xq)[2]) {
#pragma unroll
  for (int kk = 0; kk < 16; ++kk) at[kk] = *(const v2f*)(dptr + 4 * kk);
#pragma unroll
  for (int ks = 0; ks < 4; ++ks) {
    aa[ks].x = aptr[ks * 64];      // alpha^T[f=l16][n=k0]   (k0 = ks*4 + 2*hl)
    aa[ks].y = aptr[ks * 64 + 16]; // alpha^T[f=l16][n=k0+1]
  }
  if (PRE) {
    xq[0] = *(const v4f*)(xptr);
    xq[1] = *(const v4f*)(xptr + 4);
  }
  const size_t m = advance ? 1 : 0; // scalar; pointers advance via one v_add_nc_u64 each
  dptr += m * DSTRIDE;
  aptr += m * ASTRIDE;
  xptr += m * XSTRIDE;
}

// Process one chunk: cross^T via 16 WMMAs, exp2, LDS transpose, 4 WMMAs into out^T.
template <bool PRE>
__device__ __forceinline__ void do_chunk(const v2f (&at)[16], const v2f (&aa)[4],
                                         const v4f (&xq)[2], const v2f (&zt)[16],
                                         float zb, float* __restrict__ kt,
                                         int l16, int hl, v8f& oacc) {
  float xs = 0.0f;
  if (!PRE) { // fallback: row norms from the loaded tile + half swap
#pragma unroll
    for (int kk = 0; kk < 16; ++kk)
      xs = fmaf(at[kk].x, at[kk].x, fmaf(at[kk].y, at[kk].y, xs));
    xs += __shfl_xor(xs, 16, 32);
  }

  // base_r = -0.5*log2e*(zs + xsq[n0 + r + 8*hl]); independent of WMMA results,
  // so it overlaps the matrix pipe and leaves only fma+exp2 on the critical path.
  float base[8];
#pragma unroll
  for (int r = 0; r < 8; ++r) {
    float xsr;
    if (PRE)
      xsr = (r < 4) ? xq[0][r] : xq[1][r - 4];
    else
      xsr = __shfl(xs, hl ? (r + 8) : r, 32);
    base[r] = fmaf(xsr, NHALF_LOG2E, zb);
  }

  // cross^T[n][b], K-dim = 64 in 16 steps of 4
  v8f cr = {};
#pragma unroll
  for (int kk = 0; kk < 16; ++kk)
    cr = __builtin_amdgcn_wmma_f32_16x16x4_f32(false, at[kk], false, zt[kk],
                                               (short)0, cr, false, false);

  // K = exp(-sqdist/2) = 2^(cr*log2e + base); cr[r] = cross^T[n = r+8*hl][b = l16]
#pragma unroll
  for (int r = 0; r < 8; ++r)
    kt[(r + 8 * hl) * 16 + l16] = EXP2F(fmaf(cr[r], LOG2E, base[r])); // K^T[n][b]

  // WMMA#2: out^T += alpha^T(16f x 4n) * K^T(4n x 16b), 4 k-steps
#pragma unroll
  for (int ks = 0; ks < 4; ++ks) {
    const int k0 = ks * 4 + 2 * hl;
    v2f bK;
    bK.x = kt[k0 * 16 + l16]; // K^T[n=k0][b=l16] (DS in-order per wave)
    bK.y = kt[(k0 + 1) * 16 + l16];
    oacc = __builtin_amdgcn_wmma_f32_16x16x4_f32(false, aa[ks], false, bK,
                                                 (short)0, oacc, false, false);
  }
}

template <bool PRE>
__global__ __launch_bounds__(256) void rbf_fused(
    const float* __restrict__ z, const float* __restrict__ ds,
    const float* __restrict__ alpha, const float* __restrict__ xsq,
    float* __restrict__ out, float* __restrict__ partial, int B, int N) {
  __shared__ float lds_kt[WPB * 256];
  const int lane = threadIdx.x & 31;
  const int wave = threadIdx.x >> 5;
  const int l16 = lane & 15;
  const int hl = lane >> 4; // lane half: selects K{0,1} vs K{2,3} of WMMA layouts
  const int mtiles = B >> 4;
  // wave index is uniform; force it scalar so loop control / addressing go SALU
  const int gw = __builtin_amdgcn_readfirstlane(blockIdx.x * WPB + wave);
  const int mt = gw % mtiles;
  const int slice = gw / mtiles;
  const int b0 = mt << 4;
  float* kt = &lds_kt[wave * 256]; // wave-private K^T staging (16x16 f32)

  // z tile, B-layout of WMMA#1 (z^T, 4x16 per k-step)
  v2f zt[16];
  {
    const float* zr = z + (size_t)(b0 + l16) * 64 + 2 * hl;
#pragma unroll
    for (int kk = 0; kk < 16; ++kk) zt[kk] = *(const v2f*)(zr + 4 * kk);
  }
  float zs = 0.0f;
#pragma unroll
  for (int kk = 0; kk < 16; ++kk)
    zs = fmaf(zt[kk].x, zt[kk].x, fmaf(zt[kk].y, zt[kk].y, zs));
  zs += __shfl_xor(zs, 16, 32);        // ||z[b0+l16]||^2
  const float zb = zs * NHALF_LOG2E;   // loop-invariant part of the exp2 argument

  v8f oacc = {}; // out^T tile: VGPR r -> [f = r+8*hl][b = l16]

  const int nchunks = N >> 4;
  const int T = (nchunks - slice + NSLICES - 1) / NSLICES; // chunks for this wave (scalar)

  // Persistent per-lane base pointers, advanced by constant strides each prefetch.
  const float* dptr = ds + ((size_t)slice * 16 + l16) * 64 + 2 * hl;
  const float* aptr = alpha + ((size_t)slice * 16 + 2 * hl) * 16 + l16;
  const float* xptr = xsq ? (xsq + (size_t)slice * 16 + 8 * hl) : ds; // unused if !PRE

  // Software-pipelined ping-pong: prefetch chunk i+1 while computing chunk i.
  v2f atA[16], atB[16];
  v2f aaA[4], aaB[4];
  v4f xqA[2], xqB[2];
  load_chunk<PRE>(dptr, aptr, xptr, /*advance=*/(1 < T), atA, aaA, xqA); // chunk 0
  int i = 0;
  while (true) {
    load_chunk<PRE>(dptr, aptr, xptr, (i + 2 < T), atB, aaB, xqB); // chunk i+1 (or dummy)
    do_chunk<PRE>(atA, aaA, xqA, zt, zb, kt, l16, hl, oacc);
    if (++i >= T) break;

    load_chunk<PRE>(dptr, aptr, xptr, (i + 2 < T), atA, aaA, xqA);
    do_chunk<PRE>(atB, aaB, xqB, zt, zb, kt, l16, hl, oacc);
    if (++i >= T) break;
  }

  if (PRE) { // deterministic: per-slice partials, reduced by rbf_reduce
#pragma unroll
    for (int r = 0; r < 8; ++r)
      partial[((size_t)slice * B + b0 + l16) * 16 + r + 8 * hl] = oacc[r];
  } else {
#pragma unroll
    for (int r = 0; r < 8; ++r)
      atomicAdd(&out[(size_t)(b0 + l16) * 16 + r + 8 * hl], oacc[r]);
  }
}

__global__ __launch_bounds__(256) void rbf_reduce(const float* __restrict__ partial,
                                                  float* __restrict__ out, int total) {
  int i = blockIdx.x * blockDim.x + threadIdx.x;
  if (i >= total) return;
  float s = 0.0f;
#pragma unroll
  for (int sl = 0; sl < NSLICES; ++sl) s += partial[(size_t)sl * total + i];
  out[i] = s;
}

extern "C" void kernel_launch(void* const* d_in, const int* in_sizes, int n_in,
                              void* d_out, int out_size, void* d_ws, size_t ws_size,
                              hipStream_t stream) {
  const float* z = (const float*)d_in[0];     // [B, 64]
  const float* ds = (const float*)d_in[1];    // [N, 64]
  const float* alpha = (const float*)d_in[2]; // [N, 16]
  float* out = (float*)d_out;                 // [B, 16]
  const int B = in_sizes[0] / 64;             // 2048
  const int N = in_sizes[1] / 64;             // 100000

  const size_t part_sz = (size_t)NSLICES * (size_t)B * 16 * sizeof(float);
  const size_t need = part_sz + (size_t)N * sizeof(float);
  const int use_ws = (d_ws != nullptr && ws_size >= need) ? 1 : 0;

  const int mtiles = B / 16;                   // 128
  const int blocks = (mtiles * NSLICES) / WPB; // 256 blocks x 8 waves

  if (use_ws) {
    float* partial = (float*)d_ws;
    float* xsq = (float*)((char*)d_ws + part_sz);
    rbf_xsq<<<(N + 255) / 256, 256, 0, stream>>>(ds, xsq, N);
    rbf_fused<true><<<blocks, 256, 0, stream>>>(z, ds, alpha, xsq, out, partial, B, N);
    rbf_reduce<<<(out_size + 255) / 256, 256, 0, stream>>>(partial, out, out_size);
  } else {
    rbf_zero<<<(out_size + 255) / 256, 256, 0, stream>>>(out, out_size);
    rbf_fused<false><<<blocks, 256, 0, stream>>>(z, ds, alpha, nullptr, out, nullptr, B, N);
  }
}